// SineKAN_65755949302266
// MI455X (gfx1250) — compile-verified
//
#include <hip/hip_runtime.h>

#define IN_F   256
#define OUT_F  256
#define GRID_G 8
#define BATCH_N 2048

typedef __attribute__((ext_vector_type(2))) float v2f;
typedef __attribute__((ext_vector_type(4))) float v4f;
typedef __attribute__((ext_vector_type(8))) float v8f;

// ---------------------------------------------------------------------------
// Kernel 1: base = x @ W^T + bias  via V_WMMA_F32_16X16X4_F32
//   x: [2048,256] row-major (A, M x K), W: [256,256] row-major [o][i]
//   B[k=i][n=o] = W[o][i]  -> B column n is W row o (contiguous in k). [CDNA5 7.12.2]
// One 16x16 C tile per wave; 8 waves (8 tiles) per 256-thread block.
// ---------------------------------------------------------------------------
__global__ __launch_bounds__(256) void sinekan_base_wmma(
    const float* __restrict__ x, const float* __restrict__ w,
    const float* __restrict__ bias, float* __restrict__ out)
{
    const int lane = threadIdx.x & 31;
    const int wave = threadIdx.x >> 5;
    const int tile = blockIdx.x * 8 + wave;     // 2048 tiles total (128 Mt x 16 Nt)
    const int mt = tile >> 4;                   // 0..127
    const int nt = tile & 15;                   // 0..15
    const int m0 = mt * 16;
    const int n0 = nt * 16;
    const int half = lane >> 4;                 // 0 or 1
    const int l16  = lane & 15;

    // A frag (16x4 f32): lane holds x[m0+l16][k+2*half], x[m0+l16][k+2*half+1]
    // B frag (4x16 f32): lane holds W[n0+l16][k+2*half], W[n0+l16][k+2*half+1]
    const float* __restrict__ arow = x + (size_t)(m0 + l16) * IN_F;
    const float* __restrict__ brow = w + (size_t)(n0 + l16) * IN_F;

    v8f c = {};
    #pragma unroll 8
    for (int k = 0; k < IN_F; k += 4) {
        v2f a = *(const v2f*)(arow + k + 2 * half);
        v2f b = *(const v2f*)(brow + k + 2 * half);
        c = __builtin_amdgcn_wmma_f32_16x16x4_f32(
                /*neg_a=*/false, a, /*neg_b=*/false, b,
                /*c_mod=*/(short)0, c, /*reuse_a=*/false, /*reuse_b=*/false);
    }

    // C layout: VGPR r -> M = r + 8*half, N = l16 (lanes 16-31 hold M=8..15)
    const int n = n0 + l16;
    const float bv = bias[n];
    #pragma unroll
    for (int r = 0; r < 8; ++r) {
        out[(size_t)(m0 + r + 8 * half) * OUT_F + n] = c[r] + bv;
    }
}

// ---------------------------------------------------------------------------
// Kernel 2: out[n,o] += sum_{i,g} amp[o,i,g] * sin(freq[o,i,g]*x[n,i] + phase[o,i,g])
// Block tile: 32 batch rows (lanes) x 8 outputs (waves). f/p/a addresses are
// wave-uniform (o forced uniform via readfirstlane) -> broadcast b128 loads.
// x tile staged in LDS, pitch 257 floats -> bank-conflict-free per-lane reads.
// v_sin_f32 computes sin(2*pi*t), so fold 1/(2*pi) into the argument.
// ---------------------------------------------------------------------------
__global__ __launch_bounds__(256) void sinekan_sine(
    const float* __restrict__ x, const float* __restrict__ freq,
    const float* __restrict__ phase, const float* __restrict__ amp,
    float* __restrict__ out)
{
    const int lane = threadIdx.x & 31;
    const int wave = threadIdx.x >> 5;
    const int n0 = blockIdx.x * 32;                                  // batch base
    const int o  = __builtin_amdgcn_readfirstlane(blockIdx.y * 8 + wave);

    __shared__ float xs[32][IN_F + 1];   // pitch 257 -> conflict-free columns

    // Stage x[n0..n0+31][0..255]: 8192 floats, 256 threads -> 32 each, coalesced.
    for (int t = threadIdx.x; t < 32 * IN_F; t += 256) {
        const int r = t >> 8;        // / IN_F
        const int cix = t & (IN_F - 1);
        xs[r][cix] = x[(size_t)(n0 + r) * IN_F + cix];
    }
    __syncthreads();

    const v4f* __restrict__ f4 = (const v4f*)(freq  + (size_t)o * (IN_F * GRID_G));
    const v4f* __restrict__ p4 = (const v4f*)(phase + (size_t)o * (IN_F * GRID_G));
    const v4f* __restrict__ a4 = (const v4f*)(amp   + (size_t)o * (IN_F * GRID_G));

    const float inv2pi = 0.15915494309189535f;
    float acc = 0.0f;

    #pragma unroll 2
    for (int i = 0; i < IN_F; ++i) {
        const float xv = xs[lane][i];
        const v4f f0 = f4[2 * i], f1 = f4[2 * i + 1];
        const v4f p0 = p4[2 * i], p1 = p4[2 * i + 1];
        const v4f a0 = a4[2 * i], a1 = a4[2 * i + 1];
        #pragma unroll
        for (int g = 0; g < 4; ++g) {
            float t0 = __builtin_fmaf(f0[g], xv, p0[g]) * inv2pi;
            float t1 = __builtin_fmaf(f1[g], xv, p1[g]) * inv2pi;
            acc = __builtin_fmaf(a0[g], __builtin_amdgcn_sinf(t0), acc);
            acc = __builtin_fmaf(a1[g], __builtin_amdgcn_sinf(t1), acc);
        }
    }

    float* dst = out + (size_t)(n0 + lane) * OUT_F + o;
    *dst = *dst + acc;   // base kernel (same stream) already wrote base+bias
}

extern "C" void kernel_launch(void* const* d_in, const int* in_sizes, int n_in,
                              void* d_out, int out_size, void* d_ws, size_t ws_size,
                              hipStream_t stream) {
    const float* x     = (const float*)d_in[0];   // [2048,256]
    const float* freq  = (const float*)d_in[1];   // [256,256,8]
    const float* phase = (const float*)d_in[2];   // [256,256,8]
    const float* amp   = (const float*)d_in[3];   // [256,256,8]
    const float* w     = (const float*)d_in[4];   // [256,256]
    const float* bias  = (const float*)d_in[5];   // [256]
    float* out = (float*)d_out;                   // [2048,256]

    // 2048 WMMA tiles (128 x 16), 8 tiles (waves) per block -> 256 blocks
    sinekan_base_wmma<<<dim3(256), dim3(256), 0, stream>>>(x, w, bias, out);
    // 64 batch tiles x 32 output tiles
    sinekan_sine<<<dim3(64, 32), dim3(256), 0, stream>>>(x, freq, phase, amp, out);
}